// YOWO_33998961115399
// MI455X (gfx1250) — compile-verified
//
#include <hip/hip_runtime.h>
#include <cstdint>

// ---------------- problem constants ----------------
#define BATCH   64
#define NCLS    80
#define NBINS   1024     // histogram bins over p = sig(conf)*sig(cls) in (0,1)
#define CAP     2048     // candidate cap per (img,level)
#define TOPK_   300
#define NCAND   900      // 3 * 300
#define NPAD    1024     // padded NMS size (pow2)
#define PTHRESH 0.0025f  // score>0.05  <=>  p>0.0025
#define NMS_T   0.6f
#define TPB_TILES 8      // 1024-float tiles per block in streaming passes

// ---------------- CDNA5 async-to-LDS helpers ----------------
static __device__ __forceinline__ void async_copy_b128(uint32_t lds_off,
                                                       unsigned long long saddr,
                                                       int voff_bytes) {
  // gfx1250: per-lane 16B global -> LDS, tracked by ASYNCcnt (no VGPR return path)
  asm volatile("global_load_async_to_lds_b128 %0, %1, %2"
               :: "v"(lds_off), "v"(voff_bytes), "s"(saddr)
               : "memory");
}
#define ASYNC_WAIT_ALL() asm volatile("s_wait_asynccnt 0x0" ::: "memory")
#define ASYNC_WAIT_ONE() asm volatile("s_wait_asynccnt 0x1" ::: "memory")

static __device__ __forceinline__ float sigm(float x) {
  return 1.0f / (1.0f + __expf(-x));
}

// =====================================================================
// Pass A / Pass B: stream cls logits via async-to-LDS double buffering.
// PASS==0: build per-(img,lvl) histogram of p.  PASS==1: compact candidates.
// Each wave consumes exactly the LDS slice its own lanes loaded -> no barriers
// around the async pipeline; ASYNCcnt is per-wave.
// =====================================================================
template <int PASS>
__global__ __launch_bounds__(256) void score_pass_kernel(
    const float* __restrict__ conf, const float* __restrict__ cls,
    int M, int tilesPerImage, int lvl,
    uint32_t* __restrict__ hist, const int* __restrict__ cutoffBin,
    float* __restrict__ candP, uint32_t* __restrict__ candIdx,
    uint32_t* __restrict__ candCnt) {
  __shared__ float sbuf[2][1024];
  __shared__ uint32_t lh[NBINS];

  const int img = blockIdx.y;
  const int t = threadIdx.x;

  if (PASS == 0) {
    for (int b = t; b < NBINS; b += 256) lh[b] = 0u;
    __syncthreads();
  }
  int cb = 0;
  if (PASS == 1) cb = cutoffBin[img * 3 + lvl];

  const float* clsImg  = cls  + (size_t)img * M * NCLS;
  const float* confImg = conf + (size_t)img * M;

  const int tile0 = blockIdx.x * TPB_TILES;
  const int tileE = min(tile0 + TPB_TILES, tilesPerImage);
  const int nT = tileE - tile0;

  const uint32_t lds0 = (uint32_t)(uintptr_t)(&sbuf[0][t * 4]);
  const uint32_t lds1 = (uint32_t)(uintptr_t)(&sbuf[1][t * 4]);
  const unsigned long long sbase = (unsigned long long)(uintptr_t)clsImg;

  // prime the pipeline
  async_copy_b128(lds0, sbase, (tile0 * 1024 + t * 4) * 4);

  for (int k = 0; k < nT; ++k) {
    const int tile = tile0 + k;
    if (k + 1 < nT) {
      async_copy_b128(((k + 1) & 1) ? lds1 : lds0, sbase,
                      ((tile + 1) * 1024 + t * 4) * 4);
      ASYNC_WAIT_ONE();   // oldest (current tile) has landed in LDS
    } else {
      ASYNC_WAIT_ALL();
    }
    const float* buf = (k & 1) ? &sbuf[1][t * 4] : &sbuf[0][t * 4];
    const int e = tile * 1024 + t * 4;        // flat index into [M,80]
    const int m = e / NCLS;                   // groups of 4 stay in one row
    const float pc = sigm(confImg[m]);

    // Conservative group gate: p = pc*sig(cls) < pc, and bin() is monotone.
    // PASS0: if pc <= PTHRESH nothing in this group can be histogrammed.
    // PASS1: if bin(pc) < cb nothing in this group can reach the cutoff bin.
    bool groupLive;
    if (PASS == 0) {
      groupLive = (pc > PTHRESH);
    } else {
      int binc = (int)(pc * (float)NBINS);
      binc = min(binc, NBINS - 1);
      groupLive = (pc > PTHRESH) && (binc >= cb);
    }
    if (groupLive) {
#pragma unroll
      for (int q = 0; q < 4; ++q) {
        const float p = pc * sigm(buf[q]);
        if (p > PTHRESH) {
          int bin = (int)(p * (float)NBINS);
          bin = min(bin, NBINS - 1);
          if (PASS == 0) {
            atomicAdd(&lh[bin], 1u);            // ds_add
          } else if (bin >= cb) {
            const uint32_t pos = atomicAdd(&candCnt[img * 3 + lvl], 1u);
            if (pos < CAP) {
              const size_t o = (size_t)(img * 3 + lvl) * CAP + pos;
              candP[o] = p;
              candIdx[o] = (uint32_t)(e + q);
            }
          }
        }
      }
    }
  }

  if (PASS == 0) {
    __syncthreads();
    uint32_t* gh = hist + (size_t)(img * 3 + lvl) * NBINS;
    for (int b = t; b < NBINS; b += 256) {
      const uint32_t v = lh[b];
      if (v) atomicAdd(&gh[b], v);
    }
  }
}

// ---------------- cutoff: parallel suffix scan over the histogram ----------
// cb = max{b : sum_{x>=b} hist[x] >= TOPK_}, or 0 if the total is < TOPK_.
__global__ __launch_bounds__(NBINS) void cutoff_kernel(const uint32_t* __restrict__ hist,
                                                       int* __restrict__ cutoffBin) {
  __shared__ uint32_t sfx[NBINS];
  const int id = blockIdx.x;  // img*3+lvl
  const int t = threadIdx.x;
  sfx[t] = hist[(size_t)id * NBINS + t];   // coalesced, one pass
  __syncthreads();
  // Hillis-Steele suffix sums (10 steps)
  for (int off = 1; off < NBINS; off <<= 1) {
    const uint32_t v = sfx[t];
    const uint32_t a = (t + off < NBINS) ? sfx[t + off] : 0u;
    __syncthreads();
    sfx[t] = v + a;
    __syncthreads();
  }
  // sfx is non-increasing in t -> unique crossing bin
  const bool cross = (sfx[t] >= (uint32_t)TOPK_) &&
                     (t == NBINS - 1 || sfx[t + 1] < (uint32_t)TOPK_);
  if (cross) cutoffBin[id] = t;
  if (t == 0 && sfx[0] < (uint32_t)TOPK_) cutoffBin[id] = 0;
}

// ---------------- bitonic sort (descending by p, ties by idx ascending) ----
template <int N, int T>
static __device__ void bitonic_desc(float* p, uint32_t* idx) {
  const int t = threadIdx.x;
  for (int k = 2; k <= N; k <<= 1) {
    for (int j = k >> 1; j > 0; j >>= 1) {
      for (int i = t; i < N; i += T) {
        const int ixj = i ^ j;
        if (ixj > i) {
          const float pa = p[i], pb = p[ixj];
          const uint32_t ia = idx[i], ib = idx[ixj];
          const bool up = ((i & k) == 0);
          const bool aB = (pa > pb) || (pa == pb && ia < ib);
          const bool bB = (pb > pa) || (pb == pa && ib < ia);
          if (up ? bB : aB) { p[i] = pb; p[ixj] = pa; idx[i] = ib; idx[ixj] = ia; }
        }
      }
      __syncthreads();
    }
  }
}

// ---------------- per-(img,lvl): sort candidates, take 300, decode boxes ----
__global__ __launch_bounds__(1024) void topk_kernel(
    const float* __restrict__ reg, int M, int lvl, int W, float stride,
    const float* __restrict__ candP, const uint32_t* __restrict__ candIdx,
    const uint32_t* __restrict__ candCnt,
    float* __restrict__ mergedS, uint32_t* __restrict__ mergedLab,
    float* __restrict__ mergedBox) {
  __shared__ float sp[CAP];
  __shared__ uint32_t si[CAP];
  const int img = blockIdx.x;
  const int t = threadIdx.x;
  const int slot = img * 3 + lvl;
  const uint32_t cnt = candCnt[slot];
  const int n = (int)(cnt < (uint32_t)CAP ? cnt : (uint32_t)CAP);
  const float* cp = candP + (size_t)slot * CAP;
  const uint32_t* ci = candIdx + (size_t)slot * CAP;

  for (int j = t; j < CAP; j += 1024) {
    if (j < n) { sp[j] = cp[j]; si[j] = ci[j]; }
    else       { sp[j] = -1.0f; si[j] = 0xFFFFFFFFu; }
  }
  __syncthreads();
  bitonic_desc<CAP, 1024>(sp, si);

  if (t < TOPK_) {
    const float p = sp[t];
    const uint32_t idx = si[t];
    float s = 0.f, x1 = 0.f, y1 = 0.f, x2 = 0.f, y2 = 0.f;
    uint32_t lab = 0;
    if (p > 0.f) {
      const float sc = sqrtf(p);
      if (sc > 0.05f) {
        s = sc;
        lab = idx % (uint32_t)NCLS;
        const uint32_t a = idx / (uint32_t)NCLS;
        const float* r = reg + ((size_t)img * M + a) * 4;
        const float r0 = r[0], r1 = r[1], r2 = r[2], r3 = r[3];
        const float ax = (float)(a % (uint32_t)W) + 0.5f;
        const float ay = (float)(a / (uint32_t)W) + 0.5f;
        const float cx = ax * stride + r0 * stride;
        const float cy = ay * stride + r1 * stride;
        const float bw = __expf(r2) * stride;
        const float bh = __expf(r3) * stride;
        x1 = cx - 0.5f * bw; y1 = cy - 0.5f * bh;
        x2 = cx + 0.5f * bw; y2 = cy + 0.5f * bh;
      }
    }
    const int o = img * NCAND + lvl * TOPK_ + t;
    mergedS[o] = s; mergedLab[o] = lab;
    mergedBox[o * 4 + 0] = x1; mergedBox[o * 4 + 1] = y1;
    mergedBox[o * 4 + 2] = x2; mergedBox[o * 4 + 3] = y2;
  }
}

// ---------------- per-image: sort 900 by score desc (stable via index) ------
__global__ __launch_bounds__(512) void merge_sort_kernel(
    const float* __restrict__ mergedS, const uint32_t* __restrict__ mergedLab,
    const float* __restrict__ mergedBox,
    float* __restrict__ sortedS, int* __restrict__ sortedLab,
    float* __restrict__ sortedBox, float* __restrict__ sortedArea,
    uint32_t* __restrict__ liveWords) {
  __shared__ float sp[NPAD];
  __shared__ uint32_t si[NPAD];
  const int img = blockIdx.x;
  const int t = threadIdx.x;
  for (int j = t; j < NPAD; j += 512) {
    if (j < NCAND) { sp[j] = mergedS[img * NCAND + j]; si[j] = (uint32_t)j; }
    else           { sp[j] = -1.0f; si[j] = 0xFFFFFFFFu; }
  }
  __syncthreads();
  bitonic_desc<NPAD, 512>(sp, si);

  for (int j = t; j < NPAD; j += 512) {
    float s = 0.f, x1 = 0.f, y1 = 0.f, x2 = 0.f, y2 = 0.f;
    int lab = -1;
    if (j < NCAND) {  // real entries (s>=0) sort before the -1 sentinels
      const uint32_t oi = si[j];
      s = sp[j];
      lab = (int)mergedLab[img * NCAND + oi];
      x1 = mergedBox[(img * NCAND + oi) * 4 + 0];
      y1 = mergedBox[(img * NCAND + oi) * 4 + 1];
      x2 = mergedBox[(img * NCAND + oi) * 4 + 2];
      y2 = mergedBox[(img * NCAND + oi) * 4 + 3];
    }
    const int o = img * NPAD + j;
    sortedS[o] = s; sortedLab[o] = lab;
    sortedBox[o * 4 + 0] = x1; sortedBox[o * 4 + 1] = y1;
    sortedBox[o * 4 + 2] = x2; sortedBox[o * 4 + 3] = y2;
    sortedArea[o] = (x2 - x1) * (y2 - y1);
  }
  __syncthreads();
  if (t < 32) {
    uint32_t w = 0;
    for (int b = 0; b < 32; ++b) {
      const int j = t * 32 + b;
      if (j < NCAND && sp[j] > 0.f) w |= (1u << b);
    }
    liveWords[img * 32 + t] = w;
  }
}

// ---------------- bit-packed suppression matrix (parallel IoU) -------------
__global__ __launch_bounds__(256) void supmat_kernel(
    const float* __restrict__ sortedS, const int* __restrict__ sortedLab,
    const float* __restrict__ sortedBox, const float* __restrict__ sortedArea,
    uint32_t* __restrict__ sup, uint32_t* __restrict__ rowAny) {
  const int img = blockIdx.y;
  const int widx = blockIdx.x * 256 + threadIdx.x;  // row i, word w
  if (widx >= NCAND * 32) return;
  const int i = widx >> 5;
  const int w = widx & 31;
  const int base = img * NPAD;
  uint32_t word = 0;
  const float si_ = sortedS[base + i];
  if (si_ > 0.f) {
    const int li = sortedLab[base + i];
    const float ix1 = sortedBox[(base + i) * 4 + 0];
    const float iy1 = sortedBox[(base + i) * 4 + 1];
    const float ix2 = sortedBox[(base + i) * 4 + 2];
    const float iy2 = sortedBox[(base + i) * 4 + 3];
    const float ai = sortedArea[base + i];
    for (int b = 0; b < 32; ++b) {
      const int j = w * 32 + b;
      if (j < NCAND && j > i && sortedLab[base + j] == li) {
        const float x1 = fmaxf(ix1, sortedBox[(base + j) * 4 + 0]);
        const float y1 = fmaxf(iy1, sortedBox[(base + j) * 4 + 1]);
        const float x2 = fminf(ix2, sortedBox[(base + j) * 4 + 2]);
        const float y2 = fminf(iy2, sortedBox[(base + j) * 4 + 3]);
        const float inter = fmaxf(x2 - x1, 0.f) * fmaxf(y2 - y1, 0.f);
        const float iou = inter / (ai + sortedArea[base + j] - inter + 1e-10f);
        if (iou > NMS_T) word |= (1u << b);
      }
    }
  }
  sup[(size_t)img * (NCAND * 32) + widx] = word;
  if (word) atomicOr(&rowAny[img * 32 + (i >> 5)], 1u << (i & 31));
}

// ---------------- sequential greedy scan: one wave32 per image -------------
__global__ __launch_bounds__(32) void nms_scan_kernel(
    const float* __restrict__ sortedS, const float* __restrict__ sortedBox,
    const uint32_t* __restrict__ sup, const uint32_t* __restrict__ rowAny,
    const uint32_t* __restrict__ liveWords, float* __restrict__ out) {
  const int img = blockIdx.x;
  const int lane = threadIdx.x;           // wave32: lane l holds bits [32l,32l+31]
  uint32_t keepW = 0xFFFFFFFFu;
  const uint32_t auxW = rowAny[img * 32 + lane];   // nonempty rows (implies s[i]>0)
  const uint32_t* srow = sup + (size_t)img * (NCAND * 32);

  for (int i = 0; i < NCAND; ++i) {
    const int w = i >> 5, b = i & 31;
    const uint32_t kv = (uint32_t)__builtin_amdgcn_readlane((int)keepW, w);
    const uint32_t av = (uint32_t)__builtin_amdgcn_readlane((int)auxW, w);
    if (((kv & av) >> b) & 1u) {
      keepW &= ~srow[i * 32 + lane];      // coalesced 128B row, uniform branch
    }
  }
  keepW &= liveWords[img * 32 + lane];

  for (int chunk = 0; chunk < 29; ++chunk) {
    const int j = chunk * 32 + lane;
    const uint32_t kv = (uint32_t)__builtin_amdgcn_readlane((int)keepW, chunk);
    if (j < NCAND) {
      const bool kp = (kv >> lane) & 1u;
      float s = 0.f, x1 = 0.f, y1 = 0.f, x2 = 0.f, y2 = 0.f;
      if (kp) {
        const int o = img * NPAD + j;
        s = sortedS[o];
        x1 = sortedBox[o * 4 + 0]; y1 = sortedBox[o * 4 + 1];
        x2 = sortedBox[o * 4 + 2]; y2 = sortedBox[o * 4 + 3];
      }
      float* op = out + ((size_t)img * NCAND + j) * 5;
      op[0] = x1; op[1] = y1; op[2] = x2; op[3] = y2; op[4] = s;
    }
  }
}

// =====================================================================
extern "C" void kernel_launch(void* const* d_in, const int* in_sizes, int n_in,
                              void* d_out, int out_size, void* d_ws, size_t ws_size,
                              hipStream_t stream) {
  (void)in_sizes; (void)n_in; (void)out_size; (void)ws_size;
  const float* conf[3] = {(const float*)d_in[0], (const float*)d_in[3], (const float*)d_in[6]};
  const float* cls[3]  = {(const float*)d_in[1], (const float*)d_in[4], (const float*)d_in[7]};
  const float* reg[3]  = {(const float*)d_in[2], (const float*)d_in[5], (const float*)d_in[8]};
  const int   Ms[3] = {25600, 6400, 1600};
  const int   Ws[3] = {160, 80, 40};
  const float St[3] = {8.f, 16.f, 32.f};

  // ---- workspace layout (≈14.6 MB) ----
  char* ws = (char*)d_ws;
  constexpr size_t HIST_OFF   = 0;
  constexpr size_t HIST_BYTES = (size_t)BATCH * 3 * NBINS * 4;
  constexpr size_t CNT_OFF    = HIST_OFF + HIST_BYTES;
  constexpr size_t CNT_BYTES  = (size_t)BATCH * 3 * 4;
  constexpr size_t RANY_OFF   = CNT_OFF + CNT_BYTES;
  constexpr size_t RANY_BYTES = (size_t)BATCH * 32 * 4;
  constexpr size_t ZERO_BYTES = RANY_OFF + RANY_BYTES;
  constexpr size_t CUT_OFF    = ZERO_BYTES;
  constexpr size_t CANDP_OFF  = CUT_OFF   + (size_t)BATCH * 3 * 4;
  constexpr size_t CANDI_OFF  = CANDP_OFF + (size_t)BATCH * 3 * CAP * 4;
  constexpr size_t MS_OFF     = CANDI_OFF + (size_t)BATCH * 3 * CAP * 4;
  constexpr size_t ML_OFF     = MS_OFF + (size_t)BATCH * NCAND * 4;
  constexpr size_t MB_OFF     = ML_OFF + (size_t)BATCH * NCAND * 4;
  constexpr size_t SS_OFF     = MB_OFF + (size_t)BATCH * NCAND * 16;
  constexpr size_t SL_OFF     = SS_OFF + (size_t)BATCH * NPAD * 4;
  constexpr size_t SB_OFF     = SL_OFF + (size_t)BATCH * NPAD * 4;
  constexpr size_t SA_OFF     = SB_OFF + (size_t)BATCH * NPAD * 16;
  constexpr size_t LIVE_OFF   = SA_OFF + (size_t)BATCH * NPAD * 4;
  constexpr size_t SUP_OFF    = LIVE_OFF + (size_t)BATCH * 32 * 4;

  uint32_t* hist    = (uint32_t*)(ws + HIST_OFF);
  uint32_t* candCnt = (uint32_t*)(ws + CNT_OFF);
  uint32_t* rowAny  = (uint32_t*)(ws + RANY_OFF);
  int*      cutBin  = (int*)     (ws + CUT_OFF);
  float*    candP   = (float*)   (ws + CANDP_OFF);
  uint32_t* candIdx = (uint32_t*)(ws + CANDI_OFF);
  float*    mergedS = (float*)   (ws + MS_OFF);
  uint32_t* mergedL = (uint32_t*)(ws + ML_OFF);
  float*    mergedB = (float*)   (ws + MB_OFF);
  float*    sortedS = (float*)   (ws + SS_OFF);
  int*      sortedL = (int*)     (ws + SL_OFF);
  float*    sortedB = (float*)   (ws + SB_OFF);
  float*    sortedA = (float*)   (ws + SA_OFF);
  uint32_t* liveW   = (uint32_t*)(ws + LIVE_OFF);
  uint32_t* sup     = (uint32_t*)(ws + SUP_OFF);

  hipMemsetAsync(ws, 0, ZERO_BYTES, stream);  // hist + counts + rowAny

  // Pass A: histograms (async-to-LDS streaming)
  for (int l = 0; l < 3; ++l) {
    const int tiles = Ms[l] * NCLS / 1024;
    dim3 grid((tiles + TPB_TILES - 1) / TPB_TILES, BATCH);
    score_pass_kernel<0><<<grid, 256, 0, stream>>>(
        conf[l], cls[l], Ms[l], tiles, l, hist, nullptr, nullptr, nullptr, nullptr);
  }
  cutoff_kernel<<<BATCH * 3, NBINS, 0, stream>>>(hist, cutBin);

  // Pass B: compact candidates (async-to-LDS streaming)
  for (int l = 0; l < 3; ++l) {
    const int tiles = Ms[l] * NCLS / 1024;
    dim3 grid((tiles + TPB_TILES - 1) / TPB_TILES, BATCH);
    score_pass_kernel<1><<<grid, 256, 0, stream>>>(
        conf[l], cls[l], Ms[l], tiles, l, nullptr, cutBin, candP, candIdx, candCnt);
  }

  // per-level top-300 + box decode
  for (int l = 0; l < 3; ++l) {
    topk_kernel<<<BATCH, 1024, 0, stream>>>(reg[l], Ms[l], l, Ws[l], St[l],
                                            candP, candIdx, candCnt,
                                            mergedS, mergedL, mergedB);
  }

  merge_sort_kernel<<<BATCH, 512, 0, stream>>>(mergedS, mergedL, mergedB,
                                               sortedS, sortedL, sortedB,
                                               sortedA, liveW);
  {
    dim3 grid((NCAND * 32 + 255) / 256, BATCH);
    supmat_kernel<<<grid, 256, 0, stream>>>(sortedS, sortedL, sortedB, sortedA,
                                            sup, rowAny);
  }
  nms_scan_kernel<<<BATCH, 32, 0, stream>>>(sortedS, sortedB, sup, rowAny,
                                            liveW, (float*)d_out);
}